// OmniMambaBlock_25125558682002
// MI455X (gfx1250) — compile-verified
//
#include <hip/hip_runtime.h>
#include <math.h>

typedef __attribute__((ext_vector_type(2))) float v2f;
typedef __attribute__((ext_vector_type(8))) float v8f;

#define BATCH 2
#define SEQL 4096
#define ROWS (BATCH*SEQL)      // 8192
#define CCH 96
#define DIN 192
#define NST 16
#define NCHUNK 8
#define CHLEN 512
#define NDIR 4

// ---- workspace layout (float offsets) ----
#define OFF_XA     0
#define OFF_MU     (OFF_XA + ROWS*CCH)
#define OFF_RS     (OFF_MU + ROWS)
#define OFF_XZ     (OFF_RS + ROWS)
#define OFF_XCONV  (OFF_XZ + ROWS*384)
#define OFF_DELTA  (OFF_XCONV + NDIR*ROWS*DIN)
#define OFF_BC     (OFF_DELTA + NDIR*ROWS*DIN)
#define OFF_WD     (OFF_BC + NDIR*ROWS*32)
#define OFF_APROD  (OFF_WD + DIN*DIN)
#define OFF_HEND   (OFF_APROD + NDIR*BATCH*DIN*NCHUNK*NST)
#define OFF_CARRY  (OFF_HEND + NDIR*BATCH*DIN*NCHUNK*NST)
#define OFF_YBUF   (OFF_CARRY + NDIR*BATCH*DIN*NCHUNK*NST)
#define OFF_YSUM   (OFF_YBUF + (size_t)NDIR*ROWS*DIN)

// sequence index -> spatial raster index, per direction
__device__ __forceinline__ int smap(int dir, int l) {
  switch (dir) {
    case 0: return l;
    case 1: return 4095 - l;
    case 2: return ((l & 63) << 6) | (l >> 6);
    default: { int l2 = 4095 - l; return ((l2 & 63) << 6) | (l2 >> 6); }
  }
}

__device__ __forceinline__ v8f wmma4(v2f a, v2f b, v8f c) {
  // V_WMMA_F32_16X16X4_F32 : D = A(16x4) * B(4x16) + C(16x16)
  return __builtin_amdgcn_wmma_f32_16x16x4_f32(false, a, false, b, (short)0, c,
                                               false, false);
}

// ---------------- 1x1 align conv: xa[r][c] = x[b,:,p] . align_w[c,:] + b ----
__global__ void k_align(const float* __restrict__ x, const float* __restrict__ w,
                        const float* __restrict__ bias, float* __restrict__ xa) {
  int wv = threadIdx.x >> 5, lane = threadIdx.x & 31;
  int wid = blockIdx.x * 4 + wv;          // 3072 waves
  int rt = wid / 6, ct = wid % 6;
  int r0 = rt * 16, c0 = ct * 16;
  int b = r0 >> 12, p0 = r0 & 4095;
  int half = lane >> 4, ml = lane & 15;
  v8f acc = {};
  for (int k0 = 0; k0 < 64; k0 += 4) {
    int k = k0 + half * 2;
    v2f a, bv;
    int p = p0 + ml;
    a.x = x[(b * 64 + k) * SEQL + p];
    a.y = x[(b * 64 + k + 1) * SEQL + p];
    bv.x = w[(c0 + ml) * 64 + k];
    bv.y = w[(c0 + ml) * 64 + k + 1];
    acc = wmma4(a, bv, acc);
  }
  float bvv = bias[c0 + ml];
  for (int i = 0; i < 8; ++i)
    xa[(r0 + i + 8 * half) * CCH + c0 + ml] = acc[i] + bvv;
}

// ---------------- per-row layernorm stats ----------------------------------
__global__ void k_meanvar(const float* __restrict__ xa, float* __restrict__ mu,
                          float* __restrict__ rs) {
  int r = blockIdx.x * blockDim.x + threadIdx.x;
  if (r >= ROWS) return;
  float s = 0.f, s2 = 0.f;
  for (int k = 0; k < CCH; ++k) { float v = xa[r * CCH + k]; s += v; s2 += v * v; }
  float m = s * (1.f / CCH);
  float var = s2 * (1.f / CCH) - m * m;
  mu[r] = m;
  rs[r] = rsqrtf(var + 1e-5f);
}

// ---------------- fused LN + in_proj: xz = LN(xa) @ in_proj_w.T ------------
__global__ void k_ln_inproj(const float* __restrict__ xa, const float* __restrict__ mu,
                            const float* __restrict__ rs, const float* __restrict__ g,
                            const float* __restrict__ bb, const float* __restrict__ ipw,
                            float* __restrict__ xz) {
  int wv = threadIdx.x >> 5, lane = threadIdx.x & 31;
  int wid = blockIdx.x * 4 + wv;          // 12288 waves
  int rt = wid / 24, ct = wid % 24;
  int r0 = rt * 16, c0 = ct * 16;
  int half = lane >> 4, ml = lane & 15;
  int m = r0 + ml;
  float mm = mu[m], rr = rs[m];
  v8f acc = {};
  for (int k0 = 0; k0 < CCH; k0 += 4) {
    int k = k0 + half * 2;
    v2f a, bv;
    a.x = (xa[m * CCH + k] - mm) * rr * g[k] + bb[k];
    a.y = (xa[m * CCH + k + 1] - mm) * rr * g[k + 1] + bb[k + 1];
    bv.x = ipw[(c0 + ml) * CCH + k];
    bv.y = ipw[(c0 + ml) * CCH + k + 1];
    acc = wmma4(a, bv, acc);
  }
  for (int i = 0; i < 8; ++i)
    xz[(r0 + i + 8 * half) * 384 + c0 + ml] = acc[i];
}

// ---------------- W_delta = dt_proj_w(192x6) @ x_proj_w[:6](6x192) ---------
__global__ void k_wdelta(const float* __restrict__ dtw, const float* __restrict__ xpw,
                         float* __restrict__ wd) {
  int idx = blockIdx.x * blockDim.x + threadIdx.x;
  if (idx >= DIN * DIN) return;
  int i = idx / DIN, j = idx % DIN;
  float s = 0.f;
  for (int t = 0; t < 6; ++t) s += dtw[i * 6 + t] * xpw[t * DIN + j];
  wd[idx] = s;
}

// ---------------- causal depthwise conv + silu, per direction --------------
__global__ void k_conv(const float* __restrict__ xz, const float* __restrict__ cw,
                       const float* __restrict__ cb, float* __restrict__ xconv) {
  int d = threadIdx.x;                    // 0..191
  int lc = blockIdx.x & 255;
  int b = (blockIdx.x >> 8) & 1;
  int dir = blockIdx.x >> 9;
  float w0 = cw[d * 4], w1 = cw[d * 4 + 1], w2 = cw[d * 4 + 2], w3 = cw[d * 4 + 3];
  float bias = cb[d];
  float* dst = xconv + (size_t)dir * ROWS * DIN;
  for (int l = lc * 16; l < lc * 16 + 16; ++l) {
    float acc = bias;
    float wv[4] = {w0, w1, w2, w3};
#pragma unroll
    for (int j = 0; j < 4; ++j) {
      int lj = l - 3 + j;
      if (lj >= 0) {
        int p = smap(dir, lj);
        acc += wv[j] * xz[(b * SEQL + p) * 384 + d];
      }
    }
    dst[(b * SEQL + l) * DIN + d] = acc / (1.f + __expf(-acc));  // silu
  }
}

// ---------------- delta (softplus(xconv@W_delta.T + b)) and B/C proj -------
__global__ void k_xproj(const float* __restrict__ xconv, const float* __restrict__ wd,
                        const float* __restrict__ xpw, const float* __restrict__ dtb,
                        float* __restrict__ delta, float* __restrict__ bcbuf) {
  int wv = threadIdx.x >> 5, lane = threadIdx.x & 31;
  int wid = blockIdx.x * 4 + wv;          // 28672 waves
  const int WPD = 512 * 14;
  int dir = wid / WPD;
  int rem = wid % WPD;
  int rt = rem / 14, ct = rem % 14;
  int r0 = rt * 16;
  int half = lane >> 4, ml = lane & 15;
  const float* A = xconv + (size_t)dir * ROWS * DIN;
  v8f acc = {};
  if (ct < 12) {                          // delta columns 0..191
    int c0 = ct * 16;
    for (int k0 = 0; k0 < DIN; k0 += 4) {
      int k = k0 + half * 2;
      v2f a, bv;
      a.x = A[(r0 + ml) * DIN + k];
      a.y = A[(r0 + ml) * DIN + k + 1];
      bv.x = wd[(c0 + ml) * DIN + k];
      bv.y = wd[(c0 + ml) * DIN + k + 1];
      acc = wmma4(a, bv, acc);
    }
    float bias = dtb[c0 + ml];
    float* dst = delta + (size_t)dir * ROWS * DIN;
    for (int i = 0; i < 8; ++i) {
      int r = r0 + i + 8 * half;
      float xv = acc[i] + bias;
      dst[r * DIN + c0 + ml] = (xv > 20.f) ? xv : log1pf(__expf(xv));  // softplus
    }
  } else {                                // B (16) then C (16) columns
    int j0 = (ct - 12) * 16;
    for (int k0 = 0; k0 < DIN; k0 += 4) {
      int k = k0 + half * 2;
      v2f a, bv;
      a.x = A[(r0 + ml) * DIN + k];
      a.y = A[(r0 + ml) * DIN + k + 1];
      bv.x = xpw[(6 + j0 + ml) * DIN + k];
      bv.y = xpw[(6 + j0 + ml) * DIN + k + 1];
      acc = wmma4(a, bv, acc);
    }
    float* dst = bcbuf + (size_t)dir * ROWS * 32;
    for (int i = 0; i < 8; ++i) {
      int r = r0 + i + 8 * half;
      dst[r * 32 + j0 + ml] = acc[i];
    }
  }
}

// ---------------- scan pass 1: per-chunk zero-carry scan + dA product ------
__global__ void k_scan1(const float* __restrict__ delta, const float* __restrict__ xconv,
                        const float* __restrict__ bc, const float* __restrict__ A_log,
                        float* __restrict__ aprod, float* __restrict__ hend) {
  int n = threadIdx.x & 15;
  int gid = blockIdx.x * 16 + (threadIdx.x >> 4);   // 12288 groups
  int chunk = gid & 7;
  int rest = gid >> 3;
  int d = rest % DIN;
  int bd = rest / DIN;
  int b = bd & 1, dir = bd >> 1;
  float An = -__expf(A_log[d * NST + n]);
  const float* del = delta + (size_t)dir * ROWS * DIN;
  const float* xc = xconv + (size_t)dir * ROWS * DIN;
  const float* bcm = bc + (size_t)dir * ROWS * 32;
  float h = 0.f, ap = 1.f;
  int l0 = chunk * CHLEN;
  for (int l = l0; l < l0 + CHLEN; ++l) {
    int r = b * SEQL + l;
    float dv = del[r * DIN + d];
    float u = xc[r * DIN + d];
    float bm = bcm[r * 32 + n];
    float dA = __expf(dv * An);
    h = dA * h + dv * u * bm;
    ap *= dA;
  }
  aprod[gid * 16 + n] = ap;
  hend[gid * 16 + n] = h;
}

// ---------------- scan pass 2: sequential inter-chunk carry ----------------
__global__ void k_scan2(const float* __restrict__ aprod, const float* __restrict__ hend,
                        float* __restrict__ carry) {
  int n = threadIdx.x & 15;
  int gid = blockIdx.x * 16 + (threadIdx.x >> 4);   // 1536 groups
  float c = 0.f;
  for (int ch = 0; ch < NCHUNK; ++ch) {
    int idx = (gid * NCHUNK + ch) * 16 + n;
    carry[idx] = c;
    c = aprod[idx] * c + hend[idx];
  }
}

// ---------------- scan pass 3: recompute with carry, gate, scatter ---------
__global__ void k_scan3(const float* __restrict__ delta, const float* __restrict__ xconv,
                        const float* __restrict__ bc, const float* __restrict__ xz,
                        const float* __restrict__ A_log, const float* __restrict__ Dv,
                        const float* __restrict__ carry, float* __restrict__ ybuf) {
  int n = threadIdx.x & 15;
  int gid = blockIdx.x * 16 + (threadIdx.x >> 4);
  int chunk = gid & 7;
  int rest = gid >> 3;
  int d = rest % DIN;
  int bd = rest / DIN;
  int b = bd & 1, dir = bd >> 1;
  float An = -__expf(A_log[d * NST + n]);
  const float* del = delta + (size_t)dir * ROWS * DIN;
  const float* xc = xconv + (size_t)dir * ROWS * DIN;
  const float* bcm = bc + (size_t)dir * ROWS * 32;
  float* dst = ybuf + (size_t)dir * ROWS * DIN;
  float Dd = Dv[d];
  float h = carry[gid * 16 + n];
  int l0 = chunk * CHLEN;
  for (int l = l0; l < l0 + CHLEN; ++l) {
    int r = b * SEQL + l;
    float dv = del[r * DIN + d];
    float u = xc[r * DIN + d];
    float bm = bcm[r * 32 + n];
    float cm = bcm[r * 32 + 16 + n];
    float dA = __expf(dv * An);
    h = dA * h + dv * u * bm;
    float y = h * cm;
    y += __shfl_xor(y, 8, 16);
    y += __shfl_xor(y, 4, 16);
    y += __shfl_xor(y, 2, 16);
    y += __shfl_xor(y, 1, 16);
    if (n == 0) {
      int p = smap(dir, l);
      float z = xz[(b * SEQL + p) * 384 + DIN + d];
      float sz = z / (1.f + __expf(-z));
      dst[(b * SEQL + p) * DIN + d] = (y + u * Dd) * sz;
    }
  }
}

// ---------------- sum 4 directions -----------------------------------------
__global__ void k_ysum(const float* __restrict__ ybuf, float* __restrict__ ysum) {
  int i = blockIdx.x * blockDim.x + threadIdx.x;
  if (i >= ROWS * DIN) return;
  const size_t S = (size_t)ROWS * DIN;
  ysum[i] = ybuf[i] + ybuf[i + S] + ybuf[i + 2 * S] + ybuf[i + 3 * S];
}

// ---------------- out_proj (+0.25 scale + residual) with LDS transpose -----
__global__ void k_outproj(const float* __restrict__ ysum, const float* __restrict__ w,
                          const float* __restrict__ xa, float* __restrict__ out) {
  __shared__ float tile[4][16][17];
  int wv = threadIdx.x >> 5, lane = threadIdx.x & 31;
  int wid = blockIdx.x * 4 + wv;          // 3072 waves
  int rt = wid / 6, ct = wid % 6;
  int r0 = rt * 16, c0 = ct * 16;
  int half = lane >> 4, ml = lane & 15;
  v8f acc = {};
  for (int k0 = 0; k0 < DIN; k0 += 4) {
    int k = k0 + half * 2;
    v2f a, bv;
    a.x = ysum[(r0 + ml) * DIN + k];
    a.y = ysum[(r0 + ml) * DIN + k + 1];
    bv.x = w[(c0 + ml) * DIN + k];
    bv.y = w[(c0 + ml) * DIN + k + 1];
    acc = wmma4(a, bv, acc);
  }
  int b = r0 >> 12, p0 = r0 & 4095;
  for (int i = 0; i < 8; ++i) {
    int m = i + 8 * half;
    tile[wv][m][ml] = acc[i] * 0.25f + xa[(r0 + m) * CCH + c0 + ml];
  }
  __syncthreads();
  for (int i = 0; i < 8; ++i) {
    int cl = half * 8 + i;
    out[(b * CCH + c0 + cl) * SEQL + p0 + ml] = tile[wv][ml][cl];
  }
}

extern "C" void kernel_launch(void* const* d_in, const int* in_sizes, int n_in,
                              void* d_out, int out_size, void* d_ws, size_t ws_size,
                              hipStream_t stream) {
  const float* x = (const float*)d_in[0];
  const float* align_w = (const float*)d_in[1];
  const float* align_b = (const float*)d_in[2];
  const float* ln_g = (const float*)d_in[3];
  const float* ln_b = (const float*)d_in[4];
  const float* in_proj_w = (const float*)d_in[5];
  const float* conv_w = (const float*)d_in[6];
  const float* conv_b = (const float*)d_in[7];
  const float* x_proj_w = (const float*)d_in[8];
  const float* dt_proj_w = (const float*)d_in[9];
  const float* dt_proj_b = (const float*)d_in[10];
  const float* A_log = (const float*)d_in[11];
  const float* Dvec = (const float*)d_in[12];
  const float* out_proj_w = (const float*)d_in[13];
  float* out = (float*)d_out;

  float* ws = (float*)d_ws;
  float* xa = ws + OFF_XA;
  float* mu = ws + OFF_MU;
  float* rs = ws + OFF_RS;
  float* xz = ws + OFF_XZ;
  float* xconv = ws + OFF_XCONV;
  float* delta = ws + OFF_DELTA;
  float* bc = ws + OFF_BC;
  float* wd = ws + OFF_WD;
  float* aprod = ws + OFF_APROD;
  float* hend = ws + OFF_HEND;
  float* carry = ws + OFF_CARRY;
  float* ybuf = ws + OFF_YBUF;
  float* ysum = ws + OFF_YSUM;

  k_align<<<768, 128, 0, stream>>>(x, align_w, align_b, xa);
  k_meanvar<<<32, 256, 0, stream>>>(xa, mu, rs);
  k_ln_inproj<<<3072, 128, 0, stream>>>(xa, mu, rs, ln_g, ln_b, in_proj_w, xz);
  k_wdelta<<<144, 256, 0, stream>>>(dt_proj_w, x_proj_w, wd);
  k_conv<<<2048, 192, 0, stream>>>(xz, conv_w, conv_b, xconv);
  k_xproj<<<7168, 128, 0, stream>>>(xconv, wd, x_proj_w, dt_proj_b, delta, bc);
  k_scan1<<<768, 256, 0, stream>>>(delta, xconv, bc, A_log, aprod, hend);
  k_scan2<<<96, 256, 0, stream>>>(aprod, hend, carry);
  k_scan3<<<768, 256, 0, stream>>>(delta, xconv, bc, xz, A_log, Dvec, carry, ybuf);
  k_ysum<<<6144, 256, 0, stream>>>(ybuf, ysum);
  k_outproj<<<768, 128, 0, stream>>>(ysum, out_proj_w, xa, out);
}